// SemblanceLayer_5927054868902
// MI455X (gfx1250) — compile-verified
//
#include <hip/hip_runtime.h>

typedef __attribute__((ext_vector_type(2))) float v2f;
typedef __attribute__((ext_vector_type(8))) float v8f;

#define DIM0 256
#define DIM1 256
#define DIM2 1024
#define JT   16        // j-tile per block
#define KC   128       // k-chunk per block
#define KH   160       // staged k extent: [kb-16, kb+144)
#define PX   164       // x LDS row pitch (floats); 164 mod 64 = 36 -> 16 distinct banks, 16B aligned
#define PJ   16        // transposed field pitch
#define PO   132       // output staging pitch (floats); 16B aligned rows, stride-4 banks

__launch_bounds__(256)
__global__ void semblance_kernel(const float* __restrict__ x, float* __restrict__ out)
{
    __shared__ float lx[20 * PX];   // x rows j0-2 .. j0+17, zero padded
    __shared__ float la[KH * PJ];   // s1^2, layout [k''][j]; reused as output stage [j][k]
    __shared__ float lb[KH * PJ];   // 5-tap j-sum of x^2, layout [k''][j]

    const int tid = threadIdx.x;
    const int bid = blockIdx.x;
    const int kt  = bid & 7;            // k chunk index (8)
    const int jt  = (bid >> 3) & 15;    // j tile index (16)
    const int i   = bid >> 7;           // dim0 (256)
    const int j0  = jt * JT;
    const int kb  = kt * KC;

    // ---------- Phase 1: stage x into LDS (zero-padded halo) ----------
    const float* xi = x + (size_t)i * DIM1 * DIM2;
    for (int idx = tid; idx < 20 * (KH / 4); idx += 256) {
        const int r  = idx / (KH / 4);      // 0..19  -> j = j0-2+r
        const int c4 = idx % (KH / 4);      // float4 column
        const int j  = j0 - 2 + r;
        const int k  = kb - 16 + c4 * 4;
        float4 v = make_float4(0.f, 0.f, 0.f, 0.f);
        if (j >= 0 && j < DIM1) {
            const float* p = xi + (size_t)j * DIM2;
            if (k >= 0 && k + 3 < DIM2) {
                v = *(const float4*)(p + k);
            } else {
                if (k + 0 >= 0 && k + 0 < DIM2) v.x = p[k + 0];
                if (k + 1 >= 0 && k + 1 < DIM2) v.y = p[k + 1];
                if (k + 2 >= 0 && k + 2 < DIM2) v.z = p[k + 2];
                if (k + 3 >= 0 && k + 3 < DIM2) v.w = p[k + 3];
            }
        }
        *(float4*)(&lx[r * PX + c4 * 4]) = v;   // PX*4 = 656 bytes, 16B aligned
    }
    __syncthreads();

    // ---------- Phase 2: s1^2 and sum-of-squares (5-tap along j) ----------
    {
        const int tj = tid & 15;            // output j row within tile
        const int tk = tid >> 4;            // 0..15
        #pragma unroll
        for (int it = 0; it < KH / 16; ++it) {
            const int kk = tk + it * 16;    // 0..159
            float s = 0.f, ss = 0.f;
            #pragma unroll
            for (int dj = 0; dj < 5; ++dj) {
                const float v = lx[(tj + dj) * PX + kk];
                s  += v;
                ss += v * v;
            }
            la[kk * PJ + tj] = s * s;       // numerator field
            lb[kk * PJ + tj] = ss;          // denominator field
        }
    }
    __syncthreads();

    // ---------- Phase 3: depth-20 box sums as banded GEMM on the WMMA unit ----------
    // D(16k x 16j) += A(16k x 4) * B(4 x 16j), A = 0/1 band, 10 K-steps cover k' in [km-12, km+28)
    const int wave = tid >> 5;              // 0..7 -> k subtile (km = kb + 16*wave)
    const int lane = tid & 31;
    const int m    = lane & 15;             // A row M == B col N
    const int hi   = lane >> 4;             // lane half -> K components {0,1} or {2,3}
    const int base = wave * 16 + 4;         // LDS k'' index of c = 0 (i.e. k' = km-12)

    v8f dn = {};                            // numerator accumulator
    v8f dd = {};                            // denominator accumulator
    #pragma unroll
    for (int s = 0; s < 10; ++s) {
        const int c0 = 4 * s + 2 * hi;      // band-relative k' of this lane's first K component
        // band membership: output k = km+m gathers k' = km+m-10 .. km+m+9  <=>  m+2 <= c <= m+21
        v2f a;
        a.x = (c0     >= m + 2 && c0     <= m + 21) ? 1.f : 0.f;
        a.y = (c0 + 1 >= m + 2 && c0 + 1 <= m + 21) ? 1.f : 0.f;
        const int r0 = base + c0;
        v2f bn, bd;
        bn.x = la[(r0    ) * PJ + m];
        bn.y = la[(r0 + 1) * PJ + m];
        bd.x = lb[(r0    ) * PJ + m];
        bd.y = lb[(r0 + 1) * PJ + m];
        dn = __builtin_amdgcn_wmma_f32_16x16x4_f32(false, a, false, bn, (short)0, dn, false, false);
        dd = __builtin_amdgcn_wmma_f32_16x16x4_f32(false, a, false, bd, (short)0, dd, false, false);
    }

    // ---------- Phase 4: normalize + nan_to_num, stage to LDS for coalesced stores ----------
    // D layout: VGPR v -> (M = v + 8*hi, N = lane&15). Stage as [j][k], pitch PO.
    const int j  = j0 + m;
    const int jm = (j - 2 < 0) ? 0 : (j - 2);
    const int jp = (j + 2 > DIM1 - 1) ? (DIM1 - 1) : (j + 2);
    const float norm = (float)(jp - jm + 1);

    __syncthreads();                        // all waves finished reading la/lb
    float* lo = la;                         // reuse (needs 16*PO = 2112 <= 2560 floats)
    #pragma unroll
    for (int v = 0; v < 8; ++v) {
        const int M = v + 8 * hi;           // k offset within this wave's 16-k subtile
        const float num = dn[v];
        const float den = dd[v] * norm;
        float r = num / den;
        if (r != r) r = 1.0f;                                     // nan -> FILL_VALUE
        else if (r ==  __builtin_inff()) r =  3.402823466e38f;    // +inf -> f32 max (nan_to_num default)
        else if (r == -__builtin_inff()) r = -3.402823466e38f;    // -inf -> f32 lowest
        lo[m * PO + wave * 16 + M] = r;
    }
    __syncthreads();

    // ---------- Phase 5: coalesced float4 stores (one wave per contiguous 512B j-row) ----------
    for (int f = tid; f < JT * (KC / 4); f += 256) {   // 512 float4 / 256 threads = 2 iters
        const int jr = f >> 5;              // 0..15 (KC/4 = 32 float4 per row)
        const int c4 = f & 31;
        const float4 v = *(const float4*)(&lo[jr * PO + c4 * 4]);
        float* p = out + (((size_t)i * DIM1 + (j0 + jr)) * (size_t)DIM2 + (size_t)(kb + c4 * 4));
        *(float4*)p = v;
    }
}

extern "C" void kernel_launch(void* const* d_in, const int* in_sizes, int n_in,
                              void* d_out, int out_size, void* d_ws, size_t ws_size,
                              hipStream_t stream) {
    const float* x = (const float*)d_in[0];
    float* out = (float*)d_out;
    const int blocks = DIM0 * (DIM1 / JT) * (DIM2 / KC);   // 256 * 16 * 8 = 32768
    semblance_kernel<<<blocks, 256, 0, stream>>>(x, out);
}